// GReTo_28295244546605
// MI455X (gfx1250) — compile-verified
//
#include <hip/hip_runtime.h>
#include <hip/hip_bf16.h>

// ---------------------------------------------------------------------------
// GReTo spatio-temporal GNN forward for MI455X (gfx1250, wave32, WMMA).
// All dense linear algebra goes through one generic strided WMMA-f32 GEMM
// kernel using V_WMMA_F32_16X16X4_F32 (native f32 matrix op on CDNA5).
// Each wave computes a 16x64 C strip (4 WMMA tiles) so one A fragment feeds
// 4 back-to-back v_wmma issues (2.5 loads/WMMA instead of 4). All addressing
// is base-pointer + long offsets so loads stay on the GLOBAL path (no FLAT,
// no DScnt coupling in the inner loop).
// ---------------------------------------------------------------------------

typedef float v2f __attribute__((ext_vector_type(2)));
typedef float v8f __attribute__((ext_vector_type(8)));

#define NN 32768      // nodes
#define EE 262144     // edges

// ---------------------------------------------------------------------------
// Generic strided GEMM:  C[m,n] (+)= sum_tap sum_k A[m,k,tap]*B[k,n,tap] (+bias)(relu)
//   A elem: A[a_off + m*a_rs + k*a_cs + tap*a_ts]
//   B elem: B[b_off + k*b_rs + n*b_cs + tap*b_ts]
//   C elem: C[c_off + m*c_rs + n*c_cs]
// One wave computes one 16x64 strip (4 tiles); 8 waves / 256-thread block.
// Fragment layouts per CDNA5 ISA 7.12.2 (wave32):
//   A 16x4 f32 : lane m = lane&15, holds K = 2*(lane>>4) and +1 in {v0,v1}
//   B 4x16 f32 : lane n = lane&15, holds K = 2*(lane>>4) and +1 in {v0,v1}
//   C 16x16 f32: vgpr i holds rows i (lanes 0-15) and i+8 (lanes 16-31)
// Requires: M % 16 == 0, Ncols % 64 == 0, Kd % 4 == 0 (true for all call sites).
// ---------------------------------------------------------------------------
__global__ void __launch_bounds__(256)
wmma_gemm_f32(const float* __restrict__ A, const float* __restrict__ B,
              const float* __restrict__ bias, float* __restrict__ C,
              int M, int Ncols, int Kd, int taps,
              long a_off, long a_rs, long a_cs, long a_ts,
              long b_off, long b_rs, long b_cs, long b_ts,
              long c_off, long c_rs, long c_cs,
              int accum, int relu)
{
    const int lane = threadIdx.x & 31;
    const int wave = threadIdx.x >> 5;
    const int tilesN4 = Ncols >> 6;              // 64-wide column strips
    const int tile = blockIdx.x * 8 + wave;
    if (tile >= (M >> 4) * tilesN4) return;
    const int tm = tile / tilesN4;
    const int t4 = tile - tm * tilesN4;

    const int m  = (tm << 4) + (lane & 15);      // A row for this lane
    const int nl = (t4 << 6) + (lane & 15);      // column of sub-tile j is nl + 16*j
    const int kq = (lane >> 4) << 1;             // half-wave K offset: 0 or 2
    const long crow0 = (long)(tm << 4) + (long)((lane >> 4) << 3); // C rows i / i+8

    // C offsets (kept as offsets, not pointers, to stay in the global path)
    long oc[4];
#pragma unroll
    for (int j = 0; j < 4; ++j)
        oc[j] = c_off + crow0 * c_rs + (long)(nl + 16 * j) * c_cs;

    v8f c[4];
    if (accum) {
#pragma unroll
        for (int j = 0; j < 4; ++j)
#pragma unroll
            for (int i = 0; i < 8; ++i) c[j][i] = C[oc[j] + (long)i * c_rs];
    } else if (bias) {
#pragma unroll
        for (int j = 0; j < 4; ++j) {
            const float bv = bias[nl + 16 * j];
#pragma unroll
            for (int i = 0; i < 8; ++i) c[j][i] = bv;
        }
    } else {
#pragma unroll
        for (int j = 0; j < 4; ++j)
#pragma unroll
            for (int i = 0; i < 8; ++i) c[j][i] = 0.f;
    }

    for (int tap = 0; tap < taps; ++tap) {
        long oa = a_off + (long)tap * a_ts + (long)m * a_rs + (long)kq * a_cs;
        long ob[4];
#pragma unroll
        for (int j = 0; j < 4; ++j)
            ob[j] = b_off + (long)tap * b_ts + (long)(nl + 16 * j) * b_cs
                      + (long)kq * b_rs;
        const long a_step = 4 * a_cs, b_step = 4 * b_rs;

        if (a_cs == 1) {
            // contiguous K-pair in A -> single global_load_b64 per step
            for (int k0 = 0; k0 < Kd; k0 += 4) {
                v2f a = *(const v2f*)(A + oa);
#pragma unroll
                for (int j = 0; j < 4; ++j) {
                    v2f b; b.x = B[ob[j]]; b.y = B[ob[j] + b_rs];
                    c[j] = __builtin_amdgcn_wmma_f32_16x16x4_f32(
                               false, a, false, b, (short)0, c[j], false, false);
                    ob[j] += b_step;
                }
                oa += 4;
            }
        } else {
            for (int k0 = 0; k0 < Kd; k0 += 4) {
                v2f a; a.x = A[oa]; a.y = A[oa + a_cs];
#pragma unroll
                for (int j = 0; j < 4; ++j) {
                    v2f b; b.x = B[ob[j]]; b.y = B[ob[j] + b_rs];
                    c[j] = __builtin_amdgcn_wmma_f32_16x16x4_f32(
                               false, a, false, b, (short)0, c[j], false, false);
                    ob[j] += b_step;
                }
                oa += a_step;
            }
        }
    }

#pragma unroll
    for (int j = 0; j < 4; ++j)
#pragma unroll
        for (int i = 0; i < 8; ++i) {
            float v = c[j][i];
            if (relu) v = fmaxf(v, 0.f);
            C[oc[j] + (long)i * c_rs] = v;
        }
}

// ---------------------------------------------------------------------------
// Elementwise / graph kernels
// ---------------------------------------------------------------------------
__global__ void fill_f32(float* p, float v, long n) {
    long i = (long)blockIdx.x * blockDim.x + threadIdx.x;
    long st = (long)gridDim.x * blockDim.x;
    for (; i < n; i += st) p[i] = v;
}

__global__ void glu_gate(const float* __restrict__ y1, const float* __restrict__ y2,
                         float* __restrict__ o, long n) {
    long i = (long)blockIdx.x * blockDim.x + threadIdx.x;
    long st = (long)gridDim.x * blockDim.x;
    for (; i < n; i += st)
        o[i] = tanhf(y1[i]) * (1.f / (1.f + expf(-y2[i])));
}

__device__ inline float atomAddF(float* p, float v) { return unsafeAtomicAdd(p, v); }

// degrees (init to 1.0 for the self loop before calling)
__global__ void deg_accum(const int* __restrict__ dst, const float* __restrict__ ea,
                          float* __restrict__ dpos, float* __restrict__ dneg, int E) {
    int e = blockIdx.x * blockDim.x + threadIdx.x;
    if (e >= E) return;
    int d = dst[e];
    atomAddF(dpos + d, ea[2 * e] + 1.f);
    atomAddF(dneg + d, ea[2 * e + 1] + 1.f);
}

__global__ void rsqrt_inplace(float* p, long n) {
    long i = (long)blockIdx.x * blockDim.x + threadIdx.x;
    if (i < n) { float d = p[i]; p[i] = d > 0.f ? rsqrtf(d) : 0.f; }
}

__global__ void edge_norms(const int* __restrict__ src, const int* __restrict__ dst,
                           const float* __restrict__ ea,
                           const float* __restrict__ dp, const float* __restrict__ dn,
                           float* __restrict__ npos, float* __restrict__ nneg, int E) {
    int e = blockIdx.x * blockDim.x + threadIdx.x;
    if (e >= E) return;
    int s = src[e], d = dst[e];
    npos[e] = dp[s] * (ea[2 * e] + 1.f) * dp[d];
    nneg[e] = dn[s] * (ea[2 * e + 1] + 1.f) * dn[d];
}

// wave-per-edge gather/scale/scatter: agg[dst] += xw[src] * norm
__global__ void gcn_scatter(const int* __restrict__ src, const int* __restrict__ dst,
                            const float* __restrict__ norm, const float* __restrict__ xw,
                            float* __restrict__ agg, int E, int F) {
    int wid = (int)(((long)blockIdx.x * blockDim.x + threadIdx.x) >> 5);
    int lane = threadIdx.x & 31;
    if (wid >= E) return;
    long s = src[wid], d = dst[wid];           // wave-uniform -> scalar loads
    float w = norm[wid];
    long xo = s * F, ao = d * F;
    int nchunk = F >> 2;
    for (int ci = lane; ci < nchunk; ci += 32) {
        float4 v = *(const float4*)(xw + xo + 4 * ci);
        atomAddF(agg + ao + 4 * ci + 0, v.x * w);
        atomAddF(agg + ao + 4 * ci + 1, v.y * w);
        atomAddF(agg + ao + 4 * ci + 2, v.z * w);
        atomAddF(agg + ao + 4 * ci + 3, v.w * w);
    }
}

// pos = relu(agg + xw*dinv^2 (self loop) + bias); optionally acc += psi[:,k]*pos
__global__ void gcn_epilogue(const float* __restrict__ agg, const float* __restrict__ xw,
                             const float* __restrict__ dinv, const float* __restrict__ bias,
                             const float* __restrict__ psi, int kidx,
                             float* __restrict__ pos, float* __restrict__ acc,
                             long NF, int F) {
    long i = (long)blockIdx.x * blockDim.x + threadIdx.x;
    long st = (long)gridDim.x * blockDim.x;
    for (; i < NF; i += st) {
        long n = i / F;
        int f = (int)(i - n * F);
        float di = dinv[n];
        float v = agg[i] + xw[i] * di * di + bias[f];
        v = fmaxf(v, 0.f);
        pos[i] = v;
        if (acc) acc[i] += psi[n * 3 + kidx] * v;
    }
}

// psi MLP: (N,3) -> relu -> (N,64) -> (N,3)
__global__ void psi_mlp(const float* __restrict__ x, const float* __restrict__ w1,
                        const float* __restrict__ b1, const float* __restrict__ w2,
                        const float* __restrict__ b2, float* __restrict__ psi, int N) {
    int n = blockIdx.x * blockDim.x + threadIdx.x;
    if (n >= N) return;
    float a0 = x[n * 15 + 12], a1 = x[n * 15 + 13], a2 = x[n * 15 + 14];
    float p0 = b2[0], p1 = b2[1], p2 = b2[2];
#pragma unroll 4
    for (int j = 0; j < 64; ++j) {
        float h = a0 * w1[j] + a1 * w1[64 + j] + a2 * w1[128 + j] + b1[j];
        h = fmaxf(h, 0.f);
        p0 += h * w2[j * 3 + 0];
        p1 += h * w2[j * 3 + 1];
        p2 += h * w2[j * 3 + 2];
    }
    psi[n * 3 + 0] = p0; psi[n * 3 + 1] = p1; psi[n * 3 + 2] = p2;
}

// first conv: (N,12) -> (N,64,10), single input channel, ksize 3
__global__ void first_conv(const float* __restrict__ x, const float* __restrict__ w,
                           const float* __restrict__ b, float* __restrict__ out, int N) {
    int idx = blockIdx.x * blockDim.x + threadIdx.x;
    if (idx >= N * 64) return;
    int n = idx >> 6, co = idx & 63;
    const float* xr = x + (long)n * 15;
    float w0 = w[co * 3], w1 = w[co * 3 + 1], w2 = w[co * 3 + 2], bb = b[co];
    float* o = out + (long)n * 640 + co * 10;
#pragma unroll
    for (int t = 0; t < 10; ++t)
        o[t] = xr[t] * w0 + xr[t + 1] * w1 + xr[t + 2] * w2 + bb;
}

// head: relu(conv1d k=2 over (N,64,2) -> (N,128)) @ mlp_w + mlp_b ; wave per node
__global__ void head_kernel(const float* __restrict__ in, const float* __restrict__ w,
                            const float* __restrict__ b, const float* __restrict__ mw,
                            const float* __restrict__ mb, float* __restrict__ out, int N) {
    int wid = (int)(((long)blockIdx.x * blockDim.x + threadIdx.x) >> 5);
    int lane = threadIdx.x & 31;
    if (wid >= N) return;
    const float* xn = in + (long)wid * 128;
    float part = 0.f;
    for (int h = lane; h < 128; h += 32) {
        float s = b[h];
        const float* wh = w + h * 128;
#pragma unroll 8
        for (int j = 0; j < 128; ++j) s += xn[j] * wh[j];
        part += fmaxf(s, 0.f) * mw[h];
    }
#pragma unroll
    for (int off = 16; off > 0; off >>= 1) part += __shfl_down(part, off, 32);
    if (lane == 0) out[wid] = part + mb[0];
}

// ---------------------------------------------------------------------------
// Host orchestration
// ---------------------------------------------------------------------------
static inline void gemm(hipStream_t st, const float* A, const float* B, const float* bias,
                        float* C, int M, int Nc, int Kd, int taps,
                        long a_off, long a_rs, long a_cs, long a_ts,
                        long b_off, long b_rs, long b_cs, long b_ts,
                        long c_off, long c_rs, long c_cs, int accum, int relu) {
    int tiles = (M / 16) * (Nc / 64);
    int blocks = (tiles + 7) / 8;
    wmma_gemm_f32<<<blocks, 256, 0, st>>>(A, B, bias, C, M, Nc, Kd, taps,
                                          a_off, a_rs, a_cs, a_ts,
                                          b_off, b_rs, b_cs, b_ts,
                                          c_off, c_rs, c_cs, accum, relu);
}

static inline void fill(hipStream_t st, float* p, float v, long n) {
    int blocks = (int)((n + 255) / 256);
    fill_f32<<<blocks, 256, 0, st>>>(p, v, n);
}

static void run_block(hipStream_t st, const float* const* p, const float* x,
                      float* B0, float* B1, float* B2, float* B3, float* B4, float* B5,
                      float* psi, const float* dpos, const float* dneg,
                      const float* npos, const float* nneg,
                      const int* src, const int* dst, int Lin) {
    const int L1 = Lin - 2;       // tc1 output length
    const int F  = 64 * L1;       // GCN feature dim
    const long NF = (long)NN * F;
    const int eb = (NF + 255) / 256;

    // --- tc1: gated temporal conv, 3-tap shifted GEMMs, y1=B4, y2=B5 -> B1
    for (int t = 0; t < L1; ++t) {
        gemm(st, B0, p[0], p[1], B4, NN, 64, 64, 3,
             t, 64L * Lin, Lin, 1,   0, 3, 192, 1,   t, 64L * L1, L1, 0, 0);
        gemm(st, B0, p[2], p[3], B5, NN, 64, 64, 3,
             t, 64L * Lin, Lin, 1,   0, 3, 192, 1,   t, 64L * L1, L1, 0, 0);
    }
    glu_gate<<<eb, 256, 0, st>>>(B4, B5, B1, NF);

    // --- psi MLP
    psi_mlp<<<(NN + 255) / 256, 256, 0, st>>>(x, p[10], p[11], p[12], p[13], psi, NN);

    // --- K=3 positive GCN hops, accumulating acc += psi[:,i]*pos
    fill(st, B3, 0.f, NF);                 // acc
    const float* cur = B1;
    for (int i = 0; i < 3; ++i) {
        gemm(st, cur, p[4], nullptr, B4, NN, F, F, 1,
             0, F, 1, 0,   0, F, 1, 0,   0, F, 1, 0, 0);      // xw = cur @ w_pos
        fill(st, B5, 0.f, NF);
        gcn_scatter<<<(EE * 32 + 255) / 256, 256, 0, st>>>(src, dst, npos, B4, B5, EE, F);
        gcn_epilogue<<<eb, 256, 0, st>>>(B5, B4, dpos, p[5], psi, i, B2, B3, NF, F);
        cur = B2;
    }

    // --- negative GCN on out1
    gemm(st, B1, p[6], nullptr, B4, NN, F, F, 1,
         0, F, 1, 0,   0, F, 1, 0,   0, F, 1, 0, 0);
    fill(st, B5, 0.f, NF);
    gcn_scatter<<<(EE * 32 + 255) / 256, 256, 0, st>>>(src, dst, nneg, B4, B5, EE, F);
    gcn_epilogue<<<eb, 256, 0, st>>>(B5, B4, dneg, p[7], psi, 0, B2, nullptr, NF, F);

    // --- gre einsum: per-timestep (N,128)@(128,64); concat done via two halves
    for (int l = 0; l < L1; ++l) {
        gemm(st, B3, p[8], p[9], B4, NN, 64, 64, 1,               // acc half + bias
             l, 64L * L1, L1, 0,   0, 64, 1, 0,   l, 64L * L1, L1, 0, 0);
        gemm(st, B2, p[8], nullptr, B4, NN, 64, 64, 1,            // neg half, +relu
             l, 64L * L1, L1, 0,   64L * 64, 64, 1, 0,   l, 64L * L1, L1, 1, 1);
    }

    // --- tc2: gated temporal conv on B4 -> B0 (block output)
    const int L2o = L1 - 2;
    const long NF2 = (long)NN * 64 * L2o;
    for (int t = 0; t < L2o; ++t) {
        gemm(st, B4, p[14], p[15], B5, NN, 64, 64, 3,
             t, 64L * L1, L1, 1,   0, 3, 192, 1,   t, 64L * L2o, L2o, 0, 0);
        gemm(st, B4, p[16], p[17], B2, NN, 64, 64, 3,
             t, 64L * L1, L1, 1,   0, 3, 192, 1,   t, 64L * L2o, L2o, 0, 0);
    }
    glu_gate<<<(int)((NF2 + 255) / 256), 256, 0, st>>>(B5, B2, B0, NF2);
}

extern "C" void kernel_launch(void* const* d_in, const int* in_sizes, int n_in,
                              void* d_out, int out_size, void* d_ws, size_t ws_size,
                              hipStream_t stream) {
    const float* x  = (const float*)d_in[0];
    const int*   ei = (const int*)d_in[1];          // (2, E) int
    const float* ea = (const float*)d_in[2];        // (E, 2)
    const int* src = ei;
    const int* dst = ei + EE;

    // params flattened leaf-order starting at d_in[3]
    const float* P[42];
    for (int i = 0; i < 42; ++i) P[i] = (const float*)d_in[3 + i];
    const float* const* blk1 = P + 2;   // 18 leaves: tc1(w1,b1,w2,b2), w/b pos, w/b neg, gre(w,b), psi(w1,b1,w2,b2), tc2(...)
    const float* const* blk2 = P + 20;
    const float* out_w = P[38], *out_b = P[39], *mlp_w = P[40], *mlp_b = P[41];

    // workspace arena (~422 MB of f32)
    float* W = (float*)d_ws;
    float* B0 = W;  W += (long)NN * 640;
    float* B1 = W;  W += (long)NN * 512;
    float* B2 = W;  W += (long)NN * 512;
    float* B3 = W;  W += (long)NN * 512;
    float* B4 = W;  W += (long)NN * 512;
    float* B5 = W;  W += (long)NN * 512;
    float* dpos = W; W += NN;
    float* dneg = W; W += NN;
    float* npos = W; W += EE;
    float* nneg = W; W += EE;
    float* psi  = W; W += (long)NN * 3;

    // --- edge preprocessing (degrees/norms identical for both blocks)
    fill(stream, dpos, 1.f, 2L * NN);   // dpos & dneg contiguous; self loop = 1
    deg_accum<<<(EE + 255) / 256, 256, 0, stream>>>(dst, ea, dpos, dneg, EE);
    rsqrt_inplace<<<(2 * NN + 255) / 256, 256, 0, stream>>>(dpos, 2L * NN);
    edge_norms<<<(EE + 255) / 256, 256, 0, stream>>>(src, dst, ea, dpos, dneg, npos, nneg, EE);

    // --- first conv: (N,12) -> (N,64,10) in B0
    first_conv<<<(NN * 64 + 255) / 256, 256, 0, stream>>>(x, P[0], P[1], B0, NN);

    // --- two ST blocks (B0 in -> B0 out)
    run_block(stream, blk1, x, B0, B1, B2, B3, B4, B5, psi, dpos, dneg, npos, nneg, src, dst, 10);
    run_block(stream, blk2, x, B0, B1, B2, B3, B4, B5, psi, dpos, dneg, npos, nneg, src, dst, 6);

    // --- output head: (N,64,2) -> scalar per node
    head_kernel<<<(NN * 32 + 255) / 256, 256, 0, stream>>>(
        B0, out_w, out_b, mlp_w, mlp_b, (float*)d_out, NN);
}